// ClusterLinearGaussianNetwork_12704513261796
// MI455X (gfx1250) — compile-verified
//
#include <hip/hip_runtime.h>
#include <hip/hip_bf16.h>
#include <math.h>

// Problem constants (from reference)
#define NV   512
#define KC   32
#define BB   192
#define LOG2PI 1.8378770664093453f

typedef __attribute__((ext_vector_type(2))) float v2f;
typedef __attribute__((ext_vector_type(8))) float v8f;

// ---------------------------------------------------------------------------
// Kernel 1: derive assign[v] (cluster of each variable) and cluster sizes mk[k]
// ---------------------------------------------------------------------------
__global__ __launch_bounds__(NV) void prep_kernel(const float* __restrict__ C,
                                                  int* __restrict__ assign,
                                                  float* __restrict__ mk) {
    int v = threadIdx.x;                 // 0..511
    if (v < KC) mk[v] = 0.0f;
    __syncthreads();
    int a = 0;
    #pragma unroll
    for (int k = 0; k < KC; ++k)
        if (C[v * KC + k] > 0.5f) a = k;
    assign[v] = a;
    atomicAdd(&mk[a], 1.0f);
}

// ---------------------------------------------------------------------------
// Kernel 2: materialize masked weights Wm[r,c] = W[r,c] * G[assign[r],assign[c]]
// One block per row r; G row staged in LDS so the GEMM inner loop stays pure.
// ---------------------------------------------------------------------------
__global__ __launch_bounds__(256) void mask_kernel(const float* __restrict__ W,
                                                   const float* __restrict__ G,
                                                   const int* __restrict__ assign,
                                                   float* __restrict__ Wm) {
    __shared__ float grow[KC];
    const int r = blockIdx.x;            // 0..511
    const int t = threadIdx.x;
    const int aR = assign[r];
    if (t < KC) grow[t] = G[aR * KC + t];
    __syncthreads();
    const float* wr = W + r * NV;
    float*       wo = Wm + r * NV;
    for (int c = t; c < NV; c += 256)
        wo[c] = wr[c] * grow[assign[c]];
}

// ---------------------------------------------------------------------------
// Kernel 3: mean[j, r] = sum_c X[j,c] * Wm[r,c] + b[r]
// Full-precision f32 WMMA: V_WMMA_F32_16X16X4_F32, one wave per 16x16 tile.
//   M = batch rows (12 tiles), N = output vars (32 tiles), K stepped by 4.
// A layout (16x4 f32): lane&15 = M, (lane>>4)*2 = K offset, 2 VGPRs per lane.
// B layout (4x16 f32): lane&15 = N, (lane>>4)*2 = K offset.
// C/D layout (16x16 f32): lane&15 = N, VGPR q -> M = q + 8*(lane>>4).
// Dual accumulators keep two independent WMMA chains in the XDL pipe.
// ---------------------------------------------------------------------------
__global__ __launch_bounds__(128) void gemm_kernel(const float* __restrict__ X,
                                                   const float* __restrict__ Wm,
                                                   const float* __restrict__ b,
                                                   float* __restrict__ mean) {
    const int lane = threadIdx.x;                           // 0..31 (wave32)
    const int jt   = blockIdx.y * blockDim.y + threadIdx.y; // M tile 0..11
    const int rt   = blockIdx.x;                            // N tile 0..31
    const int j0   = jt * 16;
    const int r0   = rt * 16;
    const int nm   = lane & 15;
    const int half = lane >> 4;

    const float* xrow = X  + (j0 + nm) * NV + half * 2;     // A: one X row per lane
    const float* wrow = Wm + (r0 + nm) * NV + half * 2;     // B: masked-W row per lane

    v8f acc0 = {}, acc1 = {};
    #pragma unroll 4
    for (int c0 = 0; c0 < NV; c0 += 8) {
        float2 x0 = *(const float2*)(xrow + c0);
        float2 w0 = *(const float2*)(wrow + c0);
        float2 x1 = *(const float2*)(xrow + c0 + 4);
        float2 w1 = *(const float2*)(wrow + c0 + 4);
        v2f a0; a0[0] = x0.x; a0[1] = x0.y;
        v2f b0; b0[0] = w0.x; b0[1] = w0.y;
        acc0 = __builtin_amdgcn_wmma_f32_16x16x4_f32(false, a0, false, b0,
                                                     (short)0, acc0, false, false);
        v2f a1; a1[0] = x1.x; a1[1] = x1.y;
        v2f b1; b1[0] = w1.x; b1[1] = w1.y;
        acc1 = __builtin_amdgcn_wmma_f32_16x16x4_f32(false, a1, false, b1,
                                                     (short)0, acc1, false, false);
    }

    const float bias = b[r0 + nm];
    #pragma unroll
    for (int q = 0; q < 8; ++q) {
        const int M = q + 8 * half;
        mean[(j0 + M) * NV + (r0 + nm)] = acc0[q] + acc1[q] + bias;
    }
}

// ---------------------------------------------------------------------------
// Kernel 4: per-row stats for X (rows 0..191) and mean (rows 192..383):
//   rowS2[row]      = sum_v row[v]^2
//   rowCS[row*32+k] = sum_{v in cluster k} row[v]
// ---------------------------------------------------------------------------
__global__ __launch_bounds__(256) void rowstats_kernel(const float* __restrict__ X,
                                                       const float* __restrict__ mean,
                                                       const int* __restrict__ assign,
                                                       float* __restrict__ rowS2,
                                                       float* __restrict__ rowCS) {
    __shared__ float cs[KC];
    __shared__ float red[256];
    const int row = blockIdx.x;                           // 0..383
    const float* src = (row < BB) ? (X + row * NV) : (mean + (row - BB) * NV);
    const int t = threadIdx.x;
    if (t < KC) cs[t] = 0.0f;
    __syncthreads();
    float s2 = 0.0f;
    for (int v = t; v < NV; v += 256) {
        float x = src[v];
        s2 += x * x;
        atomicAdd(&cs[assign[v]], x);                     // ds_add_f32
    }
    red[t] = s2;
    __syncthreads();
    for (int off = 128; off > 0; off >>= 1) {
        if (t < off) red[t] += red[t + off];
        __syncthreads();
    }
    if (t == 0) rowS2[row] = red[0];
    if (t < KC) rowCS[row * KC + t] = cs[t];
}

// ---------------------------------------------------------------------------
// Kernel 5: column sums  u[v] = sum_i X[i,v],  w[v] = sum_j mean[j,v]
// ---------------------------------------------------------------------------
__global__ __launch_bounds__(256) void colsums_kernel(const float* __restrict__ X,
                                                      const float* __restrict__ mean,
                                                      float* __restrict__ u,
                                                      float* __restrict__ w) {
    const int v = blockIdx.x * 256 + threadIdx.x;
    if (v >= NV) return;
    float su = 0.0f, sw = 0.0f;
    for (int i = 0; i < BB; ++i) {
        su += X[i * NV + v];
        sw += mean[i * NV + v];
    }
    u[v] = su;
    w[v] = sw;
}

// ---------------------------------------------------------------------------
// Kernel 6: closed-form MVN log-prob mean via per-cluster Woodbury identity.
//   mean_ij ||X_i - mean_j||^2 = (A1+A2)/B - 2*cross/B^2
//   mean_ij (S_i^k - T_j^k)^2  = (P_k+Q_k)/B - 2*U_k*V_k/B^2
//   maha_mean = [d2 - sum_k c_k * clus_k] / (sigma^2 (1-rho)),
//       c_k = rho / (1-rho+rho*m_k)
//   logdet = n log sigma^2 + sum_k [(m_k-1)log(1-rho) + log(1-rho+rho m_k)]
// ---------------------------------------------------------------------------
__global__ __launch_bounds__(256) void final_kernel(const float* __restrict__ rowS2,
                                                    const float* __restrict__ rowCS,
                                                    const float* __restrict__ u,
                                                    const float* __restrict__ w,
                                                    const float* __restrict__ mk,
                                                    const float* __restrict__ sigma,
                                                    const float* __restrict__ rho,
                                                    float* __restrict__ out) {
    __shared__ float sA1, sA2, sCross;
    __shared__ float sU[KC], sV[KC], sP[KC], sQ[KC];
    const int t = threadIdx.x;                            // 256 threads
    if (t == 0) { sA1 = 0.0f; sA2 = 0.0f; sCross = 0.0f; }
    if (t < KC) { sU[t] = 0.0f; sV[t] = 0.0f; sP[t] = 0.0f; sQ[t] = 0.0f; }
    __syncthreads();

    float c = 0.0f;
    for (int v = t; v < NV; v += 256) c += u[v] * w[v];
    atomicAdd(&sCross, c);

    float a1 = 0.0f, a2 = 0.0f;
    for (int i = t; i < BB; i += 256) { a1 += rowS2[i]; a2 += rowS2[BB + i]; }
    atomicAdd(&sA1, a1);
    atomicAdd(&sA2, a2);

    const int k = t & 31, g = t >> 5;                     // 8 row groups x 32 clusters
    float p = 0.0f, q = 0.0f, uu = 0.0f, vv = 0.0f;
    for (int i = g; i < BB; i += 8) {
        float s  = rowCS[i * KC + k];
        float tt = rowCS[(BB + i) * KC + k];
        p += s * s; q += tt * tt; uu += s; vv += tt;
    }
    atomicAdd(&sP[k], p); atomicAdd(&sQ[k], q);
    atomicAdd(&sU[k], uu); atomicAdd(&sV[k], vv);
    __syncthreads();

    if (t == 0) {
        const float sig2 = sigma[0] * sigma[0];
        const float r1   = rho[0];
        const float omr  = 1.0f - r1;
        const float invB = 1.0f / (float)BB;
        const float d2   = (sA1 + sA2) * invB - 2.0f * sCross * invB * invB;
        float clus   = 0.0f;
        float logdet = (float)NV * logf(sig2);
        for (int kk = 0; kk < KC; ++kk) {
            const float m     = mk[kk];
            const float denom = omr + r1 * m;
            if (m > 0.5f) {
                const float ck = r1 / denom;
                clus   += ck * ((sP[kk] + sQ[kk]) * invB
                                - 2.0f * sU[kk] * sV[kk] * invB * invB);
                logdet += (m - 1.0f) * logf(omr) + logf(denom);
            }
        }
        const float maha_mean = (d2 - clus) / (sig2 * omr);
        out[0] = -0.5f * (maha_mean + logdet + (float)NV * LOG2PI);
    }
}

// ---------------------------------------------------------------------------
// Workspace layout (floats):
//   Wm | mean | assign(int) | mk | rowS2 | rowCS | u | w   (~1.5 MB total)
// ---------------------------------------------------------------------------
extern "C" void kernel_launch(void* const* d_in, const int* in_sizes, int n_in,
                              void* d_out, int out_size, void* d_ws, size_t ws_size,
                              hipStream_t stream) {
    const float* X     = (const float*)d_in[0];   // [192,512]
    const float* C     = (const float*)d_in[1];   // [512,32]
    const float* G     = (const float*)d_in[2];   // [32,32]
    const float* W     = (const float*)d_in[3];   // [512,512]
    const float* b     = (const float*)d_in[4];   // [512]
    const float* sigma = (const float*)d_in[5];   // [1]
    const float* rho   = (const float*)d_in[6];   // [1]

    float* ws     = (float*)d_ws;
    float* Wm     = ws;                           // 262144
    float* mean   = ws + 262144;                  // 98304
    int*   assign = (int*)(ws + 360448);          // 512
    float* mk     = ws + 360960;                  // 32
    float* rowS2  = ws + 360992;                  // 384
    float* rowCS  = ws + 361376;                  // 384*32 = 12288
    float* u      = ws + 373664;                  // 512
    float* w      = ws + 374176;                  // 512

    prep_kernel<<<1, NV, 0, stream>>>(C, assign, mk);
    mask_kernel<<<NV, 256, 0, stream>>>(W, G, assign, Wm);
    gemm_kernel<<<dim3(32, 3), dim3(32, 4), 0, stream>>>(X, Wm, b, mean);
    rowstats_kernel<<<2 * BB, 256, 0, stream>>>(X, mean, assign, rowS2, rowCS);
    colsums_kernel<<<2, 256, 0, stream>>>(X, mean, u, w);
    final_kernel<<<1, 256, 0, stream>>>(rowS2, rowCS, u, w, mk, sigma, rho,
                                        (float*)d_out);
}